// BlockDecomposition_7842610282510
// MI455X (gfx1250) — compile-verified
//
#include <hip/hip_runtime.h>

typedef float v2f __attribute__((ext_vector_type(2)));
typedef float v8f __attribute__((ext_vector_type(8)));

#define IDIM 128
#define NB   8
#define BS   16
#define TILE 16
#define WPB  2      // waves (tiles) per workgroup
#define XPAD 132    // 128 + 4 pad -> conflict-free column reads

// ---------------- init: zero output + relation counters ----------------
__global__ void k_init(float* __restrict__ out, int n, int* __restrict__ cnt, int nr) {
  int i = blockIdx.x * blockDim.x + threadIdx.x;
  int stride = gridDim.x * blockDim.x;
  for (int j = i; j < n; j += stride) out[j] = 0.0f;
  if (i < nr) cnt[i] = 0;
}

// ---------------- histogram of edge types (x2 for symmetrization) ------
__global__ void k_hist(const int* __restrict__ et, int E, int* __restrict__ cnt, int nr) {
  __shared__ int h[72];
  for (int j = threadIdx.x; j < nr; j += blockDim.x) h[j] = 0;
  __syncthreads();
  int i0 = blockIdx.x * blockDim.x + threadIdx.x;
  int stride = gridDim.x * blockDim.x;
  for (int i = i0; i < E; i += stride) atomicAdd(&h[et[i]], 2);
  __syncthreads();
  for (int j = threadIdx.x; j < nr; j += blockDim.x)
    if (h[j]) atomicAdd(&cnt[j], h[j]);
}

// ---------------- prefix sums: bin starts, cursors, tile starts --------
__global__ void k_prefix(const int* __restrict__ cnt, int* __restrict__ binStart,
                         int* __restrict__ cursor, int* __restrict__ tileStart, int nr) {
  if (blockIdx.x == 0 && threadIdx.x == 0) {
    int s = 0, t = 0;
    for (int r = 0; r < nr; ++r) {
      binStart[r] = s; cursor[r] = s; tileStart[r] = t;
      s += cnt[r];
      t += (cnt[r] + TILE - 1) / TILE;
    }
    binStart[nr] = s; cursor[nr] = s; tileStart[nr] = t;
  }
}

// ---------------- scatter directed edge ids into relation bins --------
__global__ void k_scatter(const int* __restrict__ et, int E,
                          int* __restrict__ cursor, int* __restrict__ bins) {
  int i0 = blockIdx.x * blockDim.x + threadIdx.x;
  int stride = gridDim.x * blockDim.x;
  for (int i = i0; i < 2 * E; i += stride) {
    int r = et[(i < E) ? i : (i - E)];
    int p = atomicAdd(&cursor[r], 1);
    bins[p] = i;
  }
}

// ---------------- main: WMMA over 16-edge same-relation tiles ---------
__global__ __launch_bounds__(WPB * 32)
void k_rgcn(const float* __restrict__ x, const float* __restrict__ bl,
            const float* __restrict__ ew, const int* __restrict__ src,
            const int* __restrict__ tgt,
            const int* __restrict__ binStart, const int* __restrict__ tileStart,
            const int* __restrict__ bins, int E, int nr, float* __restrict__ out) {
  __shared__ __align__(16) float xs[WPB][TILE][XPAD];     // w-scaled gathered x rows
  __shared__ __align__(16) float wt[WPB][NB][BS][BS];     // relation block set
  __shared__ int   m_src[WPB][TILE];
  __shared__ int   m_tgt[WPB][TILE];
  __shared__ float m_w[WPB][TILE];
  __shared__ int   t_rel[WPB];

  const int lane = threadIdx.x & 31;
  const int wid  = threadIdx.x >> 5;
  const int tile = blockIdx.x * WPB + wid;
  const int totalTiles = tileStart[nr];
  const bool active = (tile < totalTiles);

  // find this tile's relation: the r with tileStart[r] <= tile < tileStart[r+1]
  if (lane == 0) t_rel[wid] = 0;
  if (active) {
    for (int r = lane; r < nr; r += 32)
      if (tileStart[r] <= tile && tile < tileStart[r + 1]) t_rel[wid] = r;
  }
  __syncthreads();

  const int rel = t_rel[wid];
  int e0 = 0, ecnt = 0;
  if (active) {
    e0 = binStart[rel] + (tile - tileStart[rel]) * TILE;
    int e1 = binStart[rel + 1];
    int e2 = e0 + TILE;
    ecnt = ((e1 < e2) ? e1 : e2) - e0;
  }

  // per-edge metadata for this tile (w = 0 for padded rows -> zero messages)
  if (lane < TILE) {
    int sn = 0, tn = 0; float w = 0.0f;
    if (lane < ecnt) {
      int de = bins[e0 + lane];
      int e  = (de < E) ? de : (de - E);
      int s_ = src[e], t_ = tgt[e];
      sn = (de < E) ? s_ : t_;
      tn = (de < E) ? t_ : s_;
      w  = ew[e];
    }
    m_src[wid][lane] = sn;
    m_tgt[wid][lane] = tn;
    m_w[wid][lane]   = w;
  }
  __syncthreads();

  // stage gathered x rows pre-scaled by edge weight: (x*w)·W == (x·W)*w
  for (int r = 0; r < TILE; ++r) {
    float4 v = make_float4(0.0f, 0.0f, 0.0f, 0.0f);
    if (r < ecnt) {
      const float w = m_w[wid][r];
      const float4* p = (const float4*)(x + m_src[wid][r] * IDIM);
      v = p[lane];
      v.x *= w; v.y *= w; v.z *= w; v.w *= w;
    }
    *(float4*)&xs[wid][r][lane * 4] = v;
  }
  // stage relation's 8x16x16 block set (8 KB)
  {
    float4* dst = (float4*)&wt[wid][0][0][0];
    const float4* sp = (const float4*)(bl + rel * (NB * BS * BS));
    for (int i = 0; i < 16; ++i) dst[i * 32 + lane] = sp[i * 32 + lane];
  }
  __syncthreads();

  const int half = lane >> 4;   // K sub-column (ISA 16x4 A layout)
  const int nn   = lane & 15;   // A row M / B,D column N

  // hoist per-row output pointers; padded rows target out[0..127] with a 0 addend
  float* pr[8];
#pragma unroll
  for (int r = 0; r < 8; ++r)
    pr[r] = out + (m_tgt[wid][r + half * 8] * IDIM + nn);

  if (ecnt > 0) {               // wave-uniform: EXEC stays all-ones inside
#pragma unroll
    for (int b = 0; b < NB; ++b) {
      v8f c = {};
#pragma unroll
      for (int kc = 0; kc < 4; ++kc) {       // K = 16 as 4 chained K=4 WMMAs
        const int kb = kc * 4 + half * 2;
        v2f a, bb;
        a.x  = xs[wid][nn][b * BS + kb];     // A[m=nn][k], k = kb, kb+1
        a.y  = xs[wid][nn][b * BS + kb + 1];
        bb.x = wt[wid][b][kb][nn];           // B[k][n=nn]
        bb.y = wt[wid][b][kb + 1][nn];
        c = __builtin_amdgcn_wmma_f32_16x16x4_f32(
                false, a, false, bb, (short)0, c, false, false);
      }
      // D VGPR r: edge M = r + 8*half, feature N = b*16 + nn
#pragma unroll
      for (int r = 0; r < 8; ++r)
        atomicAdd(pr[r] + b * BS, c[r]);     // b*64B folds into inst offset
    }
  }
}

// ---------------- launcher ----------------
extern "C" void kernel_launch(void* const* d_in, const int* in_sizes, int n_in,
                              void* d_out, int out_size, void* d_ws, size_t ws_size,
                              hipStream_t stream) {
  const float* x  = (const float*)d_in[0];
  const float* bl = (const float*)d_in[1];
  const float* ew = (const float*)d_in[2];
  const int* src  = (const int*)d_in[3];
  const int* tgt  = (const int*)d_in[4];
  const int* et   = (const int*)d_in[5];
  float* out = (float*)d_out;

  const int E  = in_sizes[3];                    // 500000
  const int NR = in_sizes[1] / (NB * BS * BS);   // R+1 = 65 bins (covers all type ids)

  int* wsp       = (int*)d_ws;
  int* cnt       = wsp;                 // NR
  int* binStart  = cnt + NR;            // NR+1
  int* cursor    = binStart + NR + 1;   // NR+1
  int* tileStart = cursor + NR + 1;     // NR+1
  int* bins      = tileStart + NR + 1;  // 2E ints

  k_init   <<<2048, 256, 0, stream>>>(out, out_size, cnt, NR);
  k_hist   <<<1024, 256, 0, stream>>>(et, E, cnt, NR);
  k_prefix <<<1, 1, 0, stream>>>(cnt, binStart, cursor, tileStart, NR);
  k_scatter<<<2048, 256, 0, stream>>>(et, E, cursor, bins);

  const int maxTiles = (2 * E) / TILE + NR + WPB;  // upper bound on tile count
  const int nblocks  = (maxTiles + WPB - 1) / WPB;
  k_rgcn<<<nblocks, WPB * 32, 0, stream>>>(x, bl, ew, src, tgt,
                                           binStart, tileStart, bins, E, NR, out);
}